// DirectionalAttention_10110353014894
// MI455X (gfx1250) — compile-verified
//
#include <hip/hip_runtime.h>
#include <hip/hip_bf16.h>

typedef __attribute__((ext_vector_type(2))) float v2f;
typedef __attribute__((ext_vector_type(8))) float v8f;

#define Hdim 64
#define Ndim 512
#define MLPd 128
#define BPITCH 144   // m-pitch of interleaved B tile; 2*144 % 64 == 32 -> conflict-free halves

// B element (k,m) lives at ((k>>1)*BPITCH + m)*2 + (k&1); k-pairs are adjacent -> ds_load_b64
__device__ __forceinline__ int bidx(int k, int m) {
    return (((k >> 1) * BPITCH) + m) * 2 + (k & 1);
}

// ---------------------------------------------------------------------------
// Kernel 1: QKV projections.  Q/K/V[b,n,h] = feat[b,n,:]·W{q,k,v}[:,h] + b
// ---------------------------------------------------------------------------
__global__ __launch_bounds__(64) void da_qkv(
    const float* __restrict__ feat,
    const float* __restrict__ Wq, const float* __restrict__ bq,
    const float* __restrict__ Wk, const float* __restrict__ bk,
    const float* __restrict__ Wv, const float* __restrict__ bv,
    float* __restrict__ Q, float* __restrict__ K, float* __restrict__ V)
{
    __shared__ float frow[Hdim];
    const int bn = blockIdx.x;
    const int h  = threadIdx.x;
    frow[h] = feat[(size_t)bn * Hdim + h];
    __syncthreads();
    float aq = bq[h], ak = bk[h], av = bv[h];
    #pragma unroll 8
    for (int k = 0; k < Hdim; ++k) {
        const float f = frow[k];
        aq = fmaf(f, Wq[k * Hdim + h], aq);
        ak = fmaf(f, Wk[k * Hdim + h], ak);
        av = fmaf(f, Wv[k * Hdim + h], av);
    }
    Q[(size_t)bn * Hdim + h] = aq;
    K[(size_t)bn * Hdim + h] = ak;
    V[(size_t)bn * Hdim + h] = av;
}

// ---------------------------------------------------------------------------
// Kernel 2: fold Wd into W1:  Wf(4,128) = Wd(4,64) @ W1[64:128,:]
//           cvec(128) = b1 + bd @ W1[64:128,:]
// ---------------------------------------------------------------------------
__global__ __launch_bounds__(128) void da_fuse(
    const float* __restrict__ Wd, const float* __restrict__ bd,
    const float* __restrict__ W1, const float* __restrict__ b1,
    float* __restrict__ Wf, float* __restrict__ cvec)
{
    const int m = threadIdx.x;
    float a0 = 0.f, a1 = 0.f, a2 = 0.f, a3 = 0.f, c = b1[m];
    #pragma unroll 8
    for (int k = 0; k < 64; ++k) {
        const float w = W1[(Hdim + k) * MLPd + m];
        a0 = fmaf(Wd[0 * 64 + k], w, a0);
        a1 = fmaf(Wd[1 * 64 + k], w, a1);
        a2 = fmaf(Wd[2 * 64 + k], w, a2);
        a3 = fmaf(Wd[3 * 64 + k], w, a3);
        c  = fmaf(bd[k], w, c);
    }
    Wf[m] = a0; Wf[128 + m] = a1; Wf[256 + m] = a2; Wf[384 + m] = a3;
    cvec[m] = c;
}

// ---------------------------------------------------------------------------
// Kernel 3: fused directional attention, one (b,i) per 256-thread block.
//   S[j,m] = [K[b] | loc | 1 | 0](512x68) @ [Q_i*W1[:64]; Wf0; Wf1; f; 0](68x128)
//   scores[j] = sum_m relu(S[j,m]) * W2[m]    (b2 drops under softmax)
//   out[b,i,:] = softmax_j(scores) @ V[b]
// fp32 WMMA 16x16x4; B tile in LDS, k-pair interleaved for ds_load_b64.
// ---------------------------------------------------------------------------
__global__ __launch_bounds__(256) void da_main(
    const float* __restrict__ loc,    // (N,2)
    const float* __restrict__ wind,   // (B,N,2)
    const float* __restrict__ W1,     // (128,128)
    const float* __restrict__ W2,     // (128,1)
    const float* __restrict__ Qw,     // (B,N,64)
    const float* __restrict__ Kw,     // (B,N,64)
    const float* __restrict__ Vw,     // (B,N,64)
    const float* __restrict__ Wf,     // (4,128)
    const float* __restrict__ cvec,   // (128)
    float* __restrict__ out)          // (B,N,64)
{
    __shared__ float lds_B[34 * BPITCH * 2];   // 68 k-rows x 128 m, interleaved
    __shared__ float lds_q[Hdim];
    __shared__ float lds_scores[Ndim];
    __shared__ float lds_red[256];
    __shared__ float lds_inv;

    const int tid = threadIdx.x;
    const int bi  = blockIdx.x;         // b*512 + i
    const int b   = bi >> 9;
    const int i   = bi & 511;

    // --- stage Q row -------------------------------------------------------
    if (tid < Hdim) lds_q[tid] = Qw[(size_t)bi * Hdim + tid];
    __syncthreads();

    // --- build extended B matrix in LDS (interleaved layout) ---------------
    for (int idx = tid; idx < Hdim * MLPd; idx += 256) {
        const int h = idx >> 7;
        const int m = idx & 127;
        lds_B[bidx(h, m)] = lds_q[h] * W1[idx];   // W1[h*128+m] == W1[idx]
    }
    if (tid < MLPd) {
        const int m = tid;
        const float wf0 = Wf[m],        wf1 = Wf[128 + m];
        const float wf2 = Wf[256 + m],  wf3 = Wf[384 + m];
        const float li0 = loc[i * 2 + 0], li1 = loc[i * 2 + 1];
        const float w0  = wind[(size_t)bi * 2 + 0];
        const float w1  = wind[(size_t)bi * 2 + 1];
        lds_B[bidx(64, m)] = wf0;                           // row 64: +loc[j,0]*Wf0
        lds_B[bidx(65, m)] = wf1;                           // row 65: +loc[j,1]*Wf1
        lds_B[bidx(66, m)] = cvec[m] - li0 * wf0 - li1 * wf1
                           + w0 * wf2 + w1 * wf3;           // row 66: * 1.0
        lds_B[bidx(67, m)] = 0.f;                           // row 67: * 0.0
    }
    __syncthreads();

    const int wave = tid >> 5;
    const int lane = tid & 31;
    const int half = lane >> 4;   // 0: K rows +0/+1, 1: K rows +2/+3
    const int lm   = lane & 15;

    // per-lane W2 values for the 8 m-tiles (N-local = lm)
    float w2v[8];
    #pragma unroll
    for (int mt = 0; mt < 8; ++mt) w2v[mt] = W2[mt * 16 + lm];

    // --- GEMM: 32 j-tiles (16 rows each) round-robin over 8 waves ----------
    for (int jt = wave; jt < 32; jt += 8) {
        const int j0   = jt * 16;
        const int jrow = j0 + lm;
        const float* Kbase = Kw + ((size_t)b * Ndim + jrow) * Hdim;
        __builtin_prefetch(Kbase, 0, 1);

        v8f acc[8];
        #pragma unroll
        for (int mt = 0; mt < 8; ++mt) acc[mt] = (v8f){};

        for (int ks = 0; ks < 16; ++ks) {
            const int k = ks * 4 + half * 2;                 // even
            const float2 kk = *(const float2*)(Kbase + k);   // global b64
            v2f a; a.x = kk.x; a.y = kk.y;
            const float* brow = &lds_B[((k >> 1) * BPITCH) * 2];
            #pragma unroll
            for (int mt = 0; mt < 8; ++mt) {
                const v2f bf = *(const v2f*)(brow + (mt * 16 + lm) * 2);  // ds b64
                acc[mt] = __builtin_amdgcn_wmma_f32_16x16x4_f32(
                    false, a, false, bf, (short)0, acc[mt], false, false);
            }
        }
        // extension step: columns 64..67 = [loc_j0, loc_j1, 1, 0]
        {
            const float l0 = loc[jrow * 2 + 0];
            const float l1 = loc[jrow * 2 + 1];
            v2f a; a.x = half ? 1.f : l0; a.y = half ? 0.f : l1;
            const int k = 64 + half * 2;                     // even
            const float* brow = &lds_B[((k >> 1) * BPITCH) * 2];
            #pragma unroll
            for (int mt = 0; mt < 8; ++mt) {
                const v2f bf = *(const v2f*)(brow + (mt * 16 + lm) * 2);
                acc[mt] = __builtin_amdgcn_wmma_f32_16x16x4_f32(
                    false, a, false, bf, (short)0, acc[mt], false, false);
            }
        }

        // --- epilogue: relu + dot with W2, reduce 16 lanes -> score[j] -----
        // C layout: VGPR v, lane l -> j = j0 + v + 8*(l>>4), m = mt*16 + (l&15)
        #pragma unroll
        for (int v = 0; v < 8; ++v) {
            float p = 0.f;
            #pragma unroll
            for (int mt = 0; mt < 8; ++mt)
                p += fmaxf(acc[mt][v], 0.f) * w2v[mt];
            p += __shfl_xor(p, 1, 32);
            p += __shfl_xor(p, 2, 32);
            p += __shfl_xor(p, 4, 32);
            p += __shfl_xor(p, 8, 32);
            if (lm == 0) lds_scores[j0 + v + half * 8] = p;
        }
    }
    __syncthreads();

    // --- softmax over 512 scores ------------------------------------------
    const float s0 = lds_scores[tid];
    const float s1 = lds_scores[tid + 256];
    lds_red[tid] = fmaxf(s0, s1);
    __syncthreads();
    for (int off = 128; off > 0; off >>= 1) {
        if (tid < off) lds_red[tid] = fmaxf(lds_red[tid], lds_red[tid + off]);
        __syncthreads();
    }
    const float gmax = lds_red[0];
    __syncthreads();
    const float e0 = __expf(s0 - gmax);
    const float e1 = __expf(s1 - gmax);
    lds_scores[tid]       = e0;
    lds_scores[tid + 256] = e1;
    lds_red[tid] = e0 + e1;
    __syncthreads();
    for (int off = 128; off > 0; off >>= 1) {
        if (tid < off) lds_red[tid] += lds_red[tid + off];
        __syncthreads();
    }
    if (tid == 0) lds_inv = 1.f / lds_red[0];
    __syncthreads();

    // --- out[b,i,h] = (1/sum) * sum_j e[j] * V[b,j,h] ----------------------
    const int h   = tid & 63;
    const int seg = tid >> 6;              // 4 segments of 128 j's
    const float* Vbase = Vw + ((size_t)b * Ndim + seg * 128) * Hdim + h;
    float accv = 0.f;
    #pragma unroll 4
    for (int jj = 0; jj < 128; ++jj)
        accv += lds_scores[seg * 128 + jj] * Vbase[(size_t)jj * Hdim];
    lds_red[tid] = accv;
    __syncthreads();
    if (tid < Hdim) {
        const float s = lds_red[tid] + lds_red[tid + 64] +
                        lds_red[tid + 128] + lds_red[tid + 192];
        out[(size_t)bi * Hdim + tid] = s * lds_inv;
    }
}

// ---------------------------------------------------------------------------
extern "C" void kernel_launch(void* const* d_in, const int* in_sizes, int n_in,
                              void* d_out, int out_size, void* d_ws, size_t ws_size,
                              hipStream_t stream) {
    const float* features = (const float*)d_in[0];   // (4,512,64)
    const float* wind     = (const float*)d_in[1];   // (4,512,2)
    const float* loc      = (const float*)d_in[2];   // (512,2)
    const float* Wq       = (const float*)d_in[3];
    const float* bq       = (const float*)d_in[4];
    const float* Wk       = (const float*)d_in[5];
    const float* bk       = (const float*)d_in[6];
    const float* Wv       = (const float*)d_in[7];
    const float* bv       = (const float*)d_in[8];
    const float* Wd       = (const float*)d_in[9];
    const float* bd       = (const float*)d_in[10];
    const float* W1       = (const float*)d_in[11];
    const float* b1       = (const float*)d_in[12];
    const float* W2       = (const float*)d_in[13];
    // d_in[14] = b2: constant shift under softmax -> unused.

    float* ws = (float*)d_ws;
    float* Q    = ws;                 // 4*512*64 = 131072 floats
    float* K    = ws + 131072;
    float* V    = ws + 262144;
    float* Wf   = ws + 393216;        // 512 floats
    float* cvec = ws + 393728;        // 128 floats
    float* out  = (float*)d_out;      // (4,512,64) fp32

    da_qkv <<<2048, 64, 0, stream>>>(features, Wq, bq, Wk, bk, Wv, bv, Q, K, V);
    da_fuse<<<1, 128, 0, stream>>>(Wd, bd, W1, b1, Wf, cvec);
    da_main<<<2048, 256, 0, stream>>>(loc, wind, W1, W2, Q, K, V, Wf, cvec, out);
}